// GATv2_55267639165049
// MI455X (gfx1250) — compile-verified
//
#include <hip/hip_runtime.h>
#include <hip/hip_bf16.h>
#include <cstdint>

#define N_NODES   50000
#define N_EDGES   1600000
#define N_GRAPHS  64
#define E_TOT     (N_EDGES + N_NODES)   // edges + self loops = 1,650,000

typedef __attribute__((ext_vector_type(2))) float v2f;
typedef __attribute__((ext_vector_type(8))) float v8f;

// ---------------------------------------------------------------------------
// float atomic-max via monotonic int mapping.
// Init pattern: memset 0xFF (signed -1 / unsigned max) — identity for both
// paths; mixed-sign interleavings converge to the true max.
// ---------------------------------------------------------------------------
__device__ __forceinline__ void atomicMaxF(float* addr, float val) {
  if (val >= 0.0f) atomicMax((int*)addr, __float_as_int(val));
  else             atomicMin((unsigned int*)addr, (unsigned int)__float_as_int(val));
}

// ---------------------------------------------------------------------------
// Tiny weight transpose: WT[n*K + k] = W[k*N + n]  (K*N <= 8192 floats)
// Makes GEMM B-fragments contiguous 8B pairs -> global_load_b64.
// ---------------------------------------------------------------------------
__global__ void transpose_w(const float* __restrict__ W,
                            float* __restrict__ WT, int K, int N) {
  const int t = blockIdx.x * blockDim.x + threadIdx.x;
  if (t >= K * N) return;
  const int k = t / N, n = t - k * N;
  WT[n * K + k] = W[t];
}

// ---------------------------------------------------------------------------
// Fused dual GEMM: outL = A @ Wl, outR = A @ Wr  (A:[M,K] row-major,
// WlT/WrT:[N,K] pre-transposed). One wave per 16x16 C tile; 4 waves/block
// cover the 4 N-tiles (N==64). V_WMMA_F32_16X16X4_F32, EXEC all-ones.
// A frag (16x4): lanes 0-15 rows, vgpr{0,1}=K{0,1}; lanes 16-31: K{2,3}.
// B frag (4x16): lanes 0-15 cols, vgpr{0,1}=K{0,1}; lanes 16-31: K{2,3}.
// C frag: col = lane&15, row = vgpr_i + 8*(lane>>4).
// All fragment loads are 8B-aligned global_load_b64.
// ---------------------------------------------------------------------------
__global__ void gemm_dual_wmma(const float* __restrict__ A,
                               const float* __restrict__ WlT,
                               const float* __restrict__ WrT,
                               float* __restrict__ outL,
                               float* __restrict__ outR,
                               int K, int N) {
  const int lane  = threadIdx.x & 31;
  const int wave  = threadIdx.x >> 5;
  const int mBase = blockIdx.x * 16;
  const int nBase = wave * 16;          // N == 64 -> always in range

  const int rc   = lane & 15;           // A row / B col within tile
  const int kgrp = (lane >> 4) * 2;     // 0 or 2

  const float* aRow  = A   + (size_t)(mBase + rc) * K + kgrp;
  const float* blRow = WlT + (size_t)(nBase + rc) * K + kgrp;
  const float* brRow = WrT + (size_t)(nBase + rc) * K + kgrp;

  v8f accL = {}; v8f accR = {};
  for (int k0 = 0; k0 < K; k0 += 4) {
    const v2f a  = *(const v2f*)(aRow  + k0);
    const v2f bl = *(const v2f*)(blRow + k0);
    const v2f br = *(const v2f*)(brRow + k0);
    accL = __builtin_amdgcn_wmma_f32_16x16x4_f32(false, a, false, bl,
                                                 (short)0, accL, false, false);
    accR = __builtin_amdgcn_wmma_f32_16x16x4_f32(false, a, false, br,
                                                 (short)0, accR, false, false);
  }
  const int mHalf = (lane >> 4) * 8;
#pragma unroll
  for (int i = 0; i < 8; ++i) {
    outL[(size_t)(mBase + mHalf + i) * N + nBase + rc] = accL[i];
    outR[(size_t)(mBase + mHalf + i) * N + nBase + rc] = accR[i];
  }
}

// ---------------------------------------------------------------------------
// Pass A: per-edge attention scores + per-dst segment max.
// One wave per edge; lane handles channel pair {2*lane, 2*lane+1} so each
// row gather is one coalesced global_load_b64 per lane.
// For H=2 (C=32): both of a lane's channels share head = lane>>4.
// ---------------------------------------------------------------------------
template <int H, int C>
__global__ void edge_scores(const float* __restrict__ xl,
                            const float* __restrict__ xr,
                            const int* __restrict__ src,
                            const int* __restrict__ dst,
                            const float* __restrict__ att,  // [H*C]
                            float* __restrict__ scores,     // [E_TOT*H]
                            float* __restrict__ maxbuf) {   // [N_NODES*H]
  constexpr int HC = H * C;               // 64
  const int lane = threadIdx.x & 31;
  const int wid  = blockIdx.x * (blockDim.x >> 5) + (threadIdx.x >> 5);
  if (wid >= E_TOT) return;

  int s, d;
  if (wid < N_EDGES) { s = src[wid]; d = dst[wid]; }
  else               { s = d = wid - N_EDGES; }

  const int c0 = lane * 2;
  const v2f el = *(const v2f*)(xl + (size_t)s * HC + c0);
  const v2f er = *(const v2f*)(xr + (size_t)d * HC + c0);
  const v2f at = *(const v2f*)(att + c0);
  float v0 = el.x + er.x; v0 = v0 > 0.0f ? v0 : 0.2f * v0;
  float v1 = el.y + er.y; v1 = v1 > 0.0f ? v1 : 0.2f * v1;
  float p  = v0 * at.x + v1 * at.y;       // lane partial (single head's worth)

  if (H == 2) {
    // reduce within each 16-lane half (head0 = lanes 0-15, head1 = 16-31)
#pragma unroll
    for (int off = 8; off > 0; off >>= 1) p += __shfl_xor(p, off, 32);
    const float r0 = __shfl(p, 0, 32);
    const float r1 = __shfl(p, 16, 32);
    if (lane == 0) {
      scores[(size_t)wid * 2 + 0] = r0;
      scores[(size_t)wid * 2 + 1] = r1;
      atomicMaxF(&maxbuf[(size_t)d * 2 + 0], r0);
      atomicMaxF(&maxbuf[(size_t)d * 2 + 1], r1);
    }
  } else {
#pragma unroll
    for (int off = 16; off > 0; off >>= 1) p += __shfl_xor(p, off, 32);
    if (lane == 0) {
      scores[wid] = p;
      atomicMaxF(&maxbuf[d], p);
    }
  }
}

// ---------------------------------------------------------------------------
// Pass B: e = exp(score - max[dst]); scores <- e; expsum[dst] += e.
// ---------------------------------------------------------------------------
template <int H>
__global__ void edge_exp(const int* __restrict__ dst,
                         float* __restrict__ scores,
                         const float* __restrict__ maxbuf,
                         float* __restrict__ expsum) {
  const int t = blockIdx.x * blockDim.x + threadIdx.x;
  if (t >= E_TOT * H) return;
  const int e = t / H;
  const int h = t - e * H;
  const int d = (e < N_EDGES) ? dst[e] : (e - N_EDGES);
  const float m  = maxbuf[(size_t)d * H + h];
  const float ex = __expf(scores[t] - m);
  scores[t] = ex;
  atomicAdd(&expsum[(size_t)d * H + h], ex);
}

// ---------------------------------------------------------------------------
// Pass C: accum[dst] += alpha * xl[src]   (alpha = e / (sum + eps))
// One wave per edge; b64 gather + 2 f32 atomics per lane, L2-resident set.
// ---------------------------------------------------------------------------
template <int H, int C>
__global__ void edge_aggregate(const float* __restrict__ xl,
                               const int* __restrict__ src,
                               const int* __restrict__ dst,
                               const float* __restrict__ scores,
                               const float* __restrict__ expsum,
                               float* __restrict__ accum) {
  constexpr int HC = H * C;
  const int lane = threadIdx.x & 31;
  const int wid  = blockIdx.x * (blockDim.x >> 5) + (threadIdx.x >> 5);
  if (wid >= E_TOT) return;

  int s, d;
  if (wid < N_EDGES) { s = src[wid]; d = dst[wid]; }
  else               { s = d = wid - N_EDGES; }

  const int h = (H == 2) ? (lane >> 4) : 0;   // head of this lane's channels
  const float a = scores[(size_t)wid * H + h] /
                  (expsum[(size_t)d * H + h] + 1e-16f);

  const int c0 = lane * 2;
  const v2f el = *(const v2f*)(xl + (size_t)s * HC + c0);
  atomicAdd(&accum[(size_t)d * HC + c0 + 0], a * el.x);
  atomicAdd(&accum[(size_t)d * HC + c0 + 1], a * el.y);
}

// ---------------------------------------------------------------------------
// Pass D: h = ELU(accum + bias)
// ---------------------------------------------------------------------------
__global__ void bias_elu(const float* __restrict__ accum,
                         const float* __restrict__ bias,
                         float* __restrict__ out, int total) {
  const int t = blockIdx.x * blockDim.x + threadIdx.x;
  if (t >= total) return;
  const float v = accum[t] + bias[t & 63];
  out[t] = v > 0.0f ? v : (__expf(v) - 1.0f);
}

// ---------------------------------------------------------------------------
// Global mean pool (accumulate) + final linear.
// ---------------------------------------------------------------------------
__global__ void pool_accum(const float* __restrict__ h,
                           const int* __restrict__ batch,
                           float* __restrict__ sums,   // [N_GRAPHS*64]
                           float* __restrict__ cnt) {  // [N_GRAPHS]
  const int t = blockIdx.x * blockDim.x + threadIdx.x;
  if (t >= N_NODES * 64) return;
  const int node = t >> 6;
  const int c    = t & 63;
  const int g    = batch[node];
  atomicAdd(&sums[(size_t)g * 64 + c], h[t]);
  if (c == 0) atomicAdd(&cnt[g], 1.0f);
}

__global__ void pool_linear(const float* __restrict__ sums,
                            const float* __restrict__ cnt,
                            const float* __restrict__ lin_w,  // [64,2]
                            const float* __restrict__ lin_b,  // [2]
                            float* __restrict__ out) {        // [N_GRAPHS,2]
  const int t = threadIdx.x;                 // 128 threads
  if (t >= N_GRAPHS * 2) return;
  const int g = t >> 1, j = t & 1;
  float c = cnt[g]; c = c > 1.0f ? c : 1.0f;
  float acc = 0.0f;
#pragma unroll
  for (int k = 0; k < 64; ++k) acc += (sums[g * 64 + k] / c) * lin_w[k * 2 + j];
  out[t] = acc + lin_b[j];
}

// ---------------------------------------------------------------------------
extern "C" void kernel_launch(void* const* d_in, const int* in_sizes, int n_in,
                              void* d_out, int out_size, void* d_ws, size_t ws_size,
                              hipStream_t stream) {
  const float* x     = (const float*)d_in[0];
  const int*   ei    = (const int*)  d_in[1];   // [2, N_EDGES]
  const int*   batch = (const int*)  d_in[2];
  const float* Wl1   = (const float*)d_in[3];
  const float* Wr1   = (const float*)d_in[4];
  const float* att1  = (const float*)d_in[5];
  const float* b1    = (const float*)d_in[6];
  const float* Wl2   = (const float*)d_in[7];
  const float* Wr2   = (const float*)d_in[8];
  const float* att2  = (const float*)d_in[9];
  const float* b2    = (const float*)d_in[10];
  const float* lin_w = (const float*)d_in[11];
  const float* lin_b = (const float*)d_in[12];
  float* out = (float*)d_out;

  const int* src = ei;
  const int* dst = ei + N_EDGES;

  // -------- workspace carve-up (floats) --------
  float* ws = (float*)d_ws;
  size_t off = 0;
  float* xl      = ws + off; off += (size_t)N_NODES * 64;   // src transform
  float* xr      = ws + off; off += (size_t)N_NODES * 64;   // dst transform
  float* scores  = ws + off; off += (size_t)E_TOT * 2;      // per-edge scores/exp
  float* maxbuf  = ws + off; off += (size_t)N_NODES * 2;    // segment max
  float* expsum  = ws + off; off += (size_t)N_NODES * 2;    // segment exp-sum
  float* accum   = ws + off; off += (size_t)N_NODES * 64;   // segment weighted sum
  float* hbuf    = ws + off; off += (size_t)N_NODES * 64;   // layer output
  float* WlT     = ws + off; off += (size_t)128 * 64;       // transposed weights
  float* WrT     = ws + off; off += (size_t)128 * 64;
  float* sums    = ws + off; off += (size_t)N_GRAPHS * 64;
  float* cnt     = ws + off; off += (size_t)N_GRAPHS;

  const int mTiles     = N_NODES / 16;                    // 3125, exact
  const int edgeWaves  = (E_TOT + 7) / 8;                 // 8 waves / 256-thr block
  const int elemBlocks = (N_NODES * 64 + 255) / 256;

  // ================= Layer 1 (H=2, C=32, K=128) =================
  hipMemsetAsync(maxbuf, 0xFF, (size_t)N_NODES * 2 * sizeof(float), stream);
  hipMemsetAsync(expsum, 0,    (size_t)N_NODES * 2 * sizeof(float), stream);
  hipMemsetAsync(accum,  0,    (size_t)N_NODES * 64 * sizeof(float), stream);

  transpose_w<<<(128 * 64 + 255) / 256, 256, 0, stream>>>(Wl1, WlT, 128, 64);
  transpose_w<<<(128 * 64 + 255) / 256, 256, 0, stream>>>(Wr1, WrT, 128, 64);
  gemm_dual_wmma<<<mTiles, 128, 0, stream>>>(x, WlT, WrT, xl, xr, 128, 64);

  edge_scores<2, 32><<<edgeWaves, 256, 0, stream>>>(xl, xr, src, dst, att1,
                                                    scores, maxbuf);
  edge_exp<2><<<(E_TOT * 2 + 255) / 256, 256, 0, stream>>>(dst, scores, maxbuf,
                                                           expsum);
  edge_aggregate<2, 32><<<edgeWaves, 256, 0, stream>>>(xl, src, dst, scores,
                                                       expsum, accum);
  bias_elu<<<elemBlocks, 256, 0, stream>>>(accum, b1, hbuf, N_NODES * 64);

  // ================= Layer 2 (H=1, C=64, K=64) =================
  hipMemsetAsync(maxbuf, 0xFF, (size_t)N_NODES * sizeof(float), stream);
  hipMemsetAsync(expsum, 0,    (size_t)N_NODES * sizeof(float), stream);
  hipMemsetAsync(accum,  0,    (size_t)N_NODES * 64 * sizeof(float), stream);

  transpose_w<<<(64 * 64 + 255) / 256, 256, 0, stream>>>(Wl2, WlT, 64, 64);
  transpose_w<<<(64 * 64 + 255) / 256, 256, 0, stream>>>(Wr2, WrT, 64, 64);
  gemm_dual_wmma<<<mTiles, 128, 0, stream>>>(hbuf, WlT, WrT, xl, xr, 64, 64);

  edge_scores<1, 64><<<edgeWaves, 256, 0, stream>>>(xl, xr, src, dst, att2,
                                                    scores, maxbuf);
  edge_exp<1><<<(E_TOT + 255) / 256, 256, 0, stream>>>(dst, scores, maxbuf,
                                                       expsum);
  edge_aggregate<1, 64><<<edgeWaves, 256, 0, stream>>>(xl, src, dst, scores,
                                                       expsum, accum);
  bias_elu<<<elemBlocks, 256, 0, stream>>>(accum, b2, hbuf, N_NODES * 64);

  // ================= Pool + linear =================
  hipMemsetAsync(sums, 0, (size_t)N_GRAPHS * 64 * sizeof(float), stream);
  hipMemsetAsync(cnt,  0, (size_t)N_GRAPHS * sizeof(float), stream);
  pool_accum<<<elemBlocks, 256, 0, stream>>>(hbuf, batch, sums, cnt);
  pool_linear<<<1, 128, 0, stream>>>(sums, cnt, lin_w, lin_b, out);
}